// QuaternionMambaBlock_45810121179667
// MI455X (gfx1250) — compile-verified
//
#include <hip/hip_runtime.h>

// Problem constants (match reference setup_inputs)
#define BQ    2
#define LQ    4096
#define DQn   512
#define SQn   4
#define CH    64                 // timesteps per scan chunk
#define NCH   (LQ / CH)          // 64 chunks
#define ST    16                 // timesteps per staged sub-tile (TDM granularity)
#define NST   (CH / ST)          // 4 sub-tiles per chunk
#define DBLK  32                 // d-channels per wave (one per lane, wave32)
#define NDBLK (DQn / DBLK)       // 16
#define WPB   2                  // waves per block
#define EPSQ  1e-6f

// ---------------------------------------------------------------- quaternions
struct Quat { float r, i, j, k; };

__device__ __forceinline__ Quat qmul(Quat a, Quat b) {
  Quat o;
  o.r = a.r*b.r - a.i*b.i - a.j*b.j - a.k*b.k;
  o.i = a.r*b.i + a.i*b.r + a.j*b.k - a.k*b.j;
  o.j = a.r*b.j - a.i*b.k + a.j*b.r + a.k*b.i;
  o.k = a.r*b.k + a.i*b.j - a.j*b.i + a.k*b.r;
  return o;
}
__device__ __forceinline__ Quat qadd(Quat a, Quat b) {
  return Quat{a.r + b.r, a.i + b.i, a.j + b.j, a.k + b.k};
}
__device__ __forceinline__ Quat qload(const float* p) {
  float4 v = *(const float4*)p;
  return Quat{v.x, v.y, v.z, v.w};
}
__device__ __forceinline__ void qstore(float* p, Quat q) {
  *(float4*)p = make_float4(q.r, q.i, q.j, q.k);
}

// Per-timestep transition: dA = (1+x) * inv(1-x), Bu = (inv(1-x)*dt) * (Bq*u),
// x = 0.5*dt*A.  Using (1+x)(1-x)^* = (1-|x|^2, 2*xv) to avoid a full quat-mul.
__device__ __forceinline__ void quat_step(const Quat* A, float dtv, Quat uq,
                                          const Quat* Bq, Quat* dA, Quat* Bu) {
  const float c = 0.5f * dtv;
#pragma unroll
  for (int s = 0; s < SQn; ++s) {
    float xr = c * A[s].r, xi = c * A[s].i, xj = c * A[s].j, xk = c * A[s].k;
    float q2 = xr*xr + xi*xi + xj*xj + xk*xk;
    float n2 = 1.0f - 2.0f*xr + q2 + EPSQ;      // |1-x|^2 + eps
    float sv = 1.0f / n2;
    dA[s].r = (1.0f - q2) * sv;
    dA[s].i = (2.0f * sv) * xi;
    dA[s].j = (2.0f * sv) * xj;
    dA[s].k = (2.0f * sv) * xk;
    float dsv = dtv * sv;                       // fold dt into inverse scale
    Quat di{(1.0f - xr) * dsv, xi * dsv, xj * dsv, xk * dsv};
    Bu[s] = qmul(di, qmul(Bq[s], uq));
  }
}

// ---------------------------------------------------------------- TDM staging
#if defined(__HIP_DEVICE_COMPILE__) && defined(__gfx1250__) && \
    __has_builtin(__builtin_amdgcn_tensor_load_to_lds) &&       \
    __has_builtin(__builtin_amdgcn_s_wait_tensorcnt)
#define USE_TDM 1
#else
#define USE_TDM 0
#endif

#if USE_TDM
typedef unsigned int v4u __attribute__((ext_vector_type(4)));
typedef int          v4i __attribute__((ext_vector_type(4)));
typedef int          v8i __attribute__((ext_vector_type(8)));

// 2D tile load via Tensor Data Mover (D# per CDNA5 ISA ch.8).
// tile0 elements per row (4B elems), tile1 rows, stride0 = row stride in elems.
// This toolchain's builtin is the 6-arg form:
//   (uint32x4 g0, int32x8 g1, int32x4 g2, int32x4 g3, int32x8 g4, i32 cpol)
__device__ __forceinline__ void tdm_load_2d(unsigned lds_addr, const void* gptr,
                                            unsigned tile0, unsigned tile1,
                                            unsigned stride0) {
  unsigned long long ga = (unsigned long long)(size_t)gptr;
  v4u g0;
  g0.x = 1u;                                        // count=1 (valid), user mode
  g0.y = lds_addr;                                  // lds_addr (bytes)
  g0.z = (unsigned)(ga & 0xFFFFFFFFull);            // global_addr[31:0]
  g0.w = (unsigned)((ga >> 32) & 0x1FFFFFFull)      // global_addr[56:32]
       | (2u << 30);                                // type=2 ("image")
  v8i g1;
  g1[0] = (int)(2u << 16);                          // data_size=2 (4 bytes)
  g1[1] = (int)((tile0 & 0xFFFFu) << 16);           // tensor_dim0[15:0]
  g1[2] = (int)((tile0 >> 16) | ((tile1 & 0xFFFFu) << 16)); // dim0 hi | dim1 lo
  g1[3] = (int)((tile1 >> 16) | (tile0 << 16));     // dim1 hi | tile_dim0
  g1[4] = (int)(tile1 & 0xFFFFu);                   // tile_dim1 (tile_dim2=0)
  g1[5] = (int)stride0;                             // tensor_dim0_stride[31:0]
  g1[6] = 0;                                        // stride0 hi | dim1_stride lo
  g1[7] = 0;
  v4i z4 = {0, 0, 0, 0};
  v8i z8 = {0, 0, 0, 0, 0, 0, 0, 0};
  __builtin_amdgcn_tensor_load_to_lds(g0, g1, z4, z4, z8, 0);
}

// LDS layout (byte offsets inside the block's shared allocation):
//   u tiles : wave*16384 + buf*8192   (16 rows * 32 lanes * 16B)
//   dt tiles: 32768 + wave*4096 + buf*2048 (16 rows * 32 lanes * 4B)
#define U_OFF(w, bf) ((unsigned)((w) * 16384u + (bf) * 8192u))
#define D_OFF(w, bf) ((unsigned)(32768u + (w) * 4096u + (bf) * 2048u))
#define LDS_BYTES 40960
#endif

// ---------------------------------------------------------------- pass 1
// Per (b, d-block, chunk): chunk summary (P,V) per s with h_out = P*h_in + V.
__global__ void __launch_bounds__(WPB * 32) qm_pass1(
    const float* __restrict__ u, const float* __restrict__ dt,
    const float* __restrict__ Ag, const float* __restrict__ Bg,
    float* __restrict__ wsP, float* __restrict__ wsV) {
  const int lane  = threadIdx.x & 31;
  const int wv    = threadIdx.x >> 5;
  const int gw    = blockIdx.x * WPB + wv;
  const int chunk = gw % NCH;
  const int dblk  = (gw / NCH) % NDBLK;
  const int b     = gw / (NCH * NDBLK);
  const int d     = dblk * DBLK + lane;
  const int l0    = chunk * CH;

  Quat A[SQn];
#pragma unroll
  for (int s = 0; s < SQn; ++s) A[s] = qload(Ag + ((size_t)d * SQn + s) * 4);

  Quat P[SQn], V[SQn];
#pragma unroll
  for (int s = 0; s < SQn; ++s) { P[s] = Quat{1, 0, 0, 0}; V[s] = Quat{0, 0, 0, 0}; }

#if USE_TDM
  __shared__ __align__(16) unsigned char lds_all[LDS_BYTES];
  const unsigned ldsbase = (unsigned)(unsigned long long)(const void*)lds_all;
  const unsigned wvu = (unsigned)__builtin_amdgcn_readfirstlane(wv);
  auto issue = [&](int st_, int bf_) {
    const size_t row = (size_t)b * LQ + (size_t)(l0 + st_ * ST);
    tdm_load_2d(ldsbase + U_OFF(wvu, bf_), u + (row * DQn + dblk * DBLK) * 4,
                DBLK * 4, ST, DQn * 4);
    tdm_load_2d(ldsbase + D_OFF(wvu, bf_), dt + row * DQn + dblk * DBLK,
                DBLK, ST, DQn);
  };
  issue(0, 0);
#endif
  for (int st = 0; st < NST; ++st) {
    const int buf = st & 1;
#if USE_TDM
    if (st + 1 < NST) { issue(st + 1, buf ^ 1); __builtin_amdgcn_s_wait_tensorcnt(2); }
    else              { __builtin_amdgcn_s_wait_tensorcnt(0); }
    __asm__ __volatile__("" ::: "memory");   // LDS written by TDM, not by C++
#endif
#pragma unroll 1
    for (int tt = 0; tt < ST; ++tt) {
      const int l = l0 + st * ST + tt;
      float dtv; Quat uq;
#if USE_TDM
      dtv = *(const float*)(lds_all + (D_OFF(wvu, buf) + (tt * DBLK + lane) * 4));
      uq  = qload((const float*)(lds_all + (U_OFF(wvu, buf) + (tt * DBLK + lane) * 16)));
#else
      const size_t off = ((size_t)b * LQ + l) * DQn + d;
      dtv = dt[off];
      uq  = qload(u + off * 4);
#endif
      Quat Bq[SQn];
#pragma unroll
      for (int s = 0; s < SQn; ++s)
        Bq[s] = qload(Bg + (((size_t)b * LQ + l) * SQn + s) * 4);
      Quat dA[SQn], Bu[SQn];
      quat_step(A, dtv, uq, Bq, dA, Bu);
#pragma unroll
      for (int s = 0; s < SQn; ++s) {
        V[s] = qadd(qmul(dA[s], V[s]), Bu[s]);
        P[s] = qmul(dA[s], P[s]);
      }
    }
  }
  const size_t base = ((((size_t)b * NDBLK + dblk) * NCH + chunk) * DBLK + lane) * SQn;
#pragma unroll
  for (int s = 0; s < SQn; ++s) {
    qstore(wsP + (base + s) * 4, P[s]);
    qstore(wsV + (base + s) * 4, V[s]);
  }
}

// ---------------------------------------------------------------- pass 2
// Scan chunk summaries per (b,d) chain -> exclusive chunk-start states h0.
__global__ void __launch_bounds__(256) qm_pass2(const float* __restrict__ wsP,
                                                const float* __restrict__ wsV,
                                                float* __restrict__ wsH0) {
  const int tid = blockIdx.x * blockDim.x + threadIdx.x;
  if (tid >= BQ * DQn) return;
  const int b = tid / DQn, d = tid % DQn;
  const int dblk = d / DBLK, lane = d % DBLK;
  Quat h[SQn];
#pragma unroll
  for (int s = 0; s < SQn; ++s) h[s] = Quat{0, 0, 0, 0};
  for (int ch = 0; ch < NCH; ++ch) {
    const size_t base = ((((size_t)b * NDBLK + dblk) * NCH + ch) * DBLK + lane) * SQn;
#pragma unroll
    for (int s = 0; s < SQn; ++s) {
      qstore(wsH0 + (base + s) * 4, h[s]);           // exclusive prefix
      Quat Pq = qload(wsP + (base + s) * 4);
      Quat Vq = qload(wsV + (base + s) * 4);
      h[s] = qadd(qmul(Pq, h[s]), Vq);
    }
  }
}

// ---------------------------------------------------------------- pass 3
// Replay each chunk from h0, emit y[b,l,d,:] = sum_s C*h.
__global__ void __launch_bounds__(WPB * 32) qm_pass3(
    const float* __restrict__ u, const float* __restrict__ dt,
    const float* __restrict__ Ag, const float* __restrict__ Bg,
    const float* __restrict__ Cg, const float* __restrict__ wsH0,
    float* __restrict__ out) {
  const int lane  = threadIdx.x & 31;
  const int wv    = threadIdx.x >> 5;
  const int gw    = blockIdx.x * WPB + wv;
  const int chunk = gw % NCH;
  const int dblk  = (gw / NCH) % NDBLK;
  const int b     = gw / (NCH * NDBLK);
  const int d     = dblk * DBLK + lane;
  const int l0    = chunk * CH;

  Quat A[SQn];
#pragma unroll
  for (int s = 0; s < SQn; ++s) A[s] = qload(Ag + ((size_t)d * SQn + s) * 4);

  Quat h[SQn];
  const size_t base = ((((size_t)b * NDBLK + dblk) * NCH + chunk) * DBLK + lane) * SQn;
#pragma unroll
  for (int s = 0; s < SQn; ++s) h[s] = qload(wsH0 + (base + s) * 4);

#if USE_TDM
  __shared__ __align__(16) unsigned char lds_all[LDS_BYTES];
  const unsigned ldsbase = (unsigned)(unsigned long long)(const void*)lds_all;
  const unsigned wvu = (unsigned)__builtin_amdgcn_readfirstlane(wv);
  auto issue = [&](int st_, int bf_) {
    const size_t row = (size_t)b * LQ + (size_t)(l0 + st_ * ST);
    tdm_load_2d(ldsbase + U_OFF(wvu, bf_), u + (row * DQn + dblk * DBLK) * 4,
                DBLK * 4, ST, DQn * 4);
    tdm_load_2d(ldsbase + D_OFF(wvu, bf_), dt + row * DQn + dblk * DBLK,
                DBLK, ST, DQn);
  };
  issue(0, 0);
#endif
  for (int st = 0; st < NST; ++st) {
    const int buf = st & 1;
#if USE_TDM
    if (st + 1 < NST) { issue(st + 1, buf ^ 1); __builtin_amdgcn_s_wait_tensorcnt(2); }
    else              { __builtin_amdgcn_s_wait_tensorcnt(0); }
    __asm__ __volatile__("" ::: "memory");
#endif
#pragma unroll 1
    for (int tt = 0; tt < ST; ++tt) {
      const int l = l0 + st * ST + tt;
      float dtv; Quat uq;
#if USE_TDM
      dtv = *(const float*)(lds_all + (D_OFF(wvu, buf) + (tt * DBLK + lane) * 4));
      uq  = qload((const float*)(lds_all + (U_OFF(wvu, buf) + (tt * DBLK + lane) * 16)));
#else
      const size_t off = ((size_t)b * LQ + l) * DQn + d;
      dtv = dt[off];
      uq  = qload(u + off * 4);
#endif
      Quat Bq[SQn];
#pragma unroll
      for (int s = 0; s < SQn; ++s)
        Bq[s] = qload(Bg + (((size_t)b * LQ + l) * SQn + s) * 4);
      Quat dA[SQn], Bu[SQn];
      quat_step(A, dtv, uq, Bq, dA, Bu);
      Quat y{0, 0, 0, 0};
#pragma unroll
      for (int s = 0; s < SQn; ++s) {
        h[s] = qadd(qmul(dA[s], h[s]), Bu[s]);
        Quat Cq = qload(Cg + (((size_t)b * LQ + l) * SQn + s) * 4);
        y = qadd(y, qmul(Cq, h[s]));
      }
      qstore(out + (((size_t)b * LQ + l) * DQn + d) * 4, y);
    }
  }
}

// ---------------------------------------------------------------- launch
extern "C" void kernel_launch(void* const* d_in, const int* in_sizes, int n_in,
                              void* d_out, int out_size, void* d_ws, size_t ws_size,
                              hipStream_t stream) {
  const float* u  = (const float*)d_in[0];
  const float* dt = (const float*)d_in[1];
  const float* Ag = (const float*)d_in[2];
  const float* Bg = (const float*)d_in[3];
  const float* Cg = (const float*)d_in[4];
  float* out = (float*)d_out;

  float* wsP = (float*)d_ws;
  const size_t per = (size_t)BQ * NDBLK * NCH * DBLK * SQn * 4; // 1,048,576 floats
  float* wsV  = wsP + per;
  float* wsH0 = wsV + per;                                      // 12 MB total

  const int nblk = (BQ * NDBLK * NCH) / WPB;                    // 1024 blocks
  qm_pass1<<<nblk, WPB * 32, 0, stream>>>(u, dt, Ag, Bg, wsP, wsV);
  qm_pass2<<<(BQ * DQn + 255) / 256, 256, 0, stream>>>(wsP, wsV, wsH0);
  qm_pass3<<<nblk, WPB * 32, 0, stream>>>(u, dt, Ag, Bg, Cg, wsH0, out);

  (void)in_sizes; (void)n_in; (void)out_size; (void)ws_size;
}